// YOLOLoss_48189533061310
// MI455X (gfx1250) — compile-verified
//
#include <hip/hip_runtime.h>
#include <math.h>

typedef float v2f __attribute__((ext_vector_type(2)));
typedef float v8f __attribute__((ext_vector_type(8)));

#define T_TGT 64
#define NUM_B 8
#define NUM_A 3
#define NC    80
#define EPSF  1e-16f
#define IOU_T 0.213f
#define IGN_T 0.7f

struct ScaleP {
    int   H, W;
    float red, xys;
    float aw[3], ah[3];
};

__device__ __forceinline__ float sigm(float x) {
    return 1.0f / (1.0f + __expf(-x));
}
// -slog(x) = min(-log(x), 100) implemented as -max(log(x), -100)
__device__ __forceinline__ float nslog(float x) {
    return -fmaxf(__logf(x), -100.0f);
}

// Full 32-lane f32 sum using V_WMMA_F32_16X16X4_F32 as a cross-lane reducer.
// A (16x4): VGPR0 holds K0 (lanes 0-15 -> M=lane) / K2 (lanes 16-31), VGPR1 = K1/K3.
// With a = {v, 0} and B = ones, D[m,n] = v[lane m] + v[lane m+16].
// D layout: VGPR j, lanes 0-15 -> M=j ; lanes 16-31 -> M=8+j. Summing the 8 D
// registers gives rows 0-7 (lanes<16) / rows 8-15 (lanes>=16); one shfl_xor(16)
// completes the wave sum on every lane.
__device__ __forceinline__ float wave_sum(float v) {
#if defined(__gfx1250__) && __has_builtin(__builtin_amdgcn_wmma_f32_16x16x4_f32)
    v2f a; a.x = v;    a.y = 0.0f;
    v2f b; b.x = 1.0f; b.y = 1.0f;
    v8f c = {};
    v8f d = __builtin_amdgcn_wmma_f32_16x16x4_f32(false, a, false, b,
                                                  (short)0, c, false, false);
    float s = d[0] + d[1] + d[2] + d[3] + d[4] + d[5] + d[6] + d[7];
    s += __shfl_xor(s, 16, 32);
    return s;
#else
    for (int off = 16; off > 0; off >>= 1) v += __shfl_xor(v, off, 32);
    return v;
#endif
}

// ---------------------------------------------------------------------------
// Negative objectness: one thread per (b, a, y, x) cell. Per-target constants
// (scaled corners under the cxcywh-misread, area, cell index, per-anchor match
// bits) are hoisted into LDS once per block.
// ---------------------------------------------------------------------------
__global__ __launch_bounds__(256)
void yolo_negobj_kernel(const float* __restrict__ pred,
                        const float* __restrict__ targets,
                        float* __restrict__ out, ScaleP sp)
{
    const int H = sp.H, W = sp.W;
    const int HW = H * W;
    const int N  = NUM_B * NUM_A * HW;

    __shared__ float s_qx1[T_TGT], s_qy1[T_TGT], s_qx2[T_TGT], s_qy2[T_TGT];
    __shared__ float s_qar[T_TGT];
    __shared__ int   s_bid[T_TGT], s_ti[T_TGT], s_tj[T_TGT], s_mb[T_TGT];

    const int tid = threadIdx.x;
    if (tid < T_TGT) {
        float x1 = targets[tid * 5 + 0] / sp.red;
        float y1 = targets[tid * 5 + 1] / sp.red;
        float x2 = targets[tid * 5 + 2] / sp.red;
        float y2 = targets[tid * 5 + 3] / sp.red;
        s_bid[tid] = (int)targets[tid * 5 + 4];
        // _pair_iou treats the xyxy target as cxcywh (reference quirk):
        s_qx1[tid] = x1 - x2 * 0.5f;
        s_qx2[tid] = x1 + x2 * 0.5f;
        s_qy1[tid] = y1 - y2 * 0.5f;
        s_qy2[tid] = y1 + y2 * 0.5f;
        s_qar[tid] = x2 * y2;
        float cx = (x1 + x2) * 0.5f, cy = (y1 + y2) * 0.5f;
        int ti = (int)floorf(cx); ti = min(max(ti, 0), W - 1);
        int tj = (int)floorf(cy); tj = min(max(tj, 0), H - 1);
        s_ti[tid] = ti; s_tj[tid] = tj;
        float tw = x2 - x1, th = y2 - y1, ta = tw * th;
        float iou3[3]; float best = -1.0f; int bi = 0;
        #pragma unroll
        for (int k = 0; k < 3; k++) {
            float inter = fminf(sp.aw[k], tw) * fminf(sp.ah[k], th);
            float iu = inter / (sp.aw[k] * sp.ah[k] + ta - inter);
            iou3[k] = iu;
            if (iu > best) { best = iu; bi = k; }
        }
        int bits = 0;
        #pragma unroll
        for (int k = 0; k < 3; k++)
            if (iou3[k] > IOU_T || k == bi) bits |= (1 << k);
        s_mb[tid] = bits;
    }
    __syncthreads();

    const int idx = blockIdx.x * blockDim.x + tid;
    float acc = 0.0f;
    if (idx < N) {
        const int x = idx % W;
        const int y = (idx / W) % H;
        const int a = (idx / HW) % NUM_A;
        const int b = idx / (HW * NUM_A);
        const float* p = pred + (b * 255 + a * 85) * HW + y * W + x;
        const float px = p[0];
        const float py = p[1 * HW];
        const float pw = p[2 * HW];
        const float ph = p[3 * HW];
        const float po = p[4 * HW];

        const float off = 0.5f * (sp.xys - 1.0f);
        const float bx = sigm(px) * sp.xys - off + (float)x;
        const float by = sigm(py) * sp.xys - off + (float)y;
        const float bw = __expf(pw) * sp.aw[a];
        const float bh = __expf(ph) * sp.ah[a];
        const float px1 = bx - bw * 0.5f, px2 = bx + bw * 0.5f;
        const float py1 = by - bh * 0.5f, py2 = by + bh * 0.5f;
        const float parea = bw * bh;

        bool ignore = false, covered = false;
        #pragma unroll 4
        for (int t = 0; t < T_TGT; t++) {
            if (s_bid[t] != b) continue;
            float ix = fminf(px2, s_qx2[t]) - fmaxf(px1, s_qx1[t]);
            float iy = fminf(py2, s_qy2[t]) - fmaxf(py1, s_qy1[t]);
            float inter = fmaxf(ix, 0.0f) * fmaxf(iy, 0.0f);
            float iou = inter / (parea + s_qar[t] + EPSF - inter);
            ignore = ignore || (iou > IGN_T);
            covered = covered ||
                      ((s_ti[t] == x) && (s_tj[t] == y) && ((s_mb[t] >> a) & 1));
        }
        if (!ignore && !covered) acc = nslog(1.0f - sigm(po));
    }

    // All 256 threads reach here uniformly -> EXEC all ones for WMMA.
    float wsum = wave_sum(acc);
    if ((tid & 31) == 0) atomicAdd(&out[1], wsum);
}

// ---------------------------------------------------------------------------
// Positive terms: one thread per (anchor, target) pair for one scale.
// ---------------------------------------------------------------------------
__global__ __launch_bounds__(256)
void yolo_pos_kernel(const float* __restrict__ pred,
                     const float* __restrict__ targets,
                     const int* __restrict__ cats,
                     float* __restrict__ out, ScaleP sp)
{
    const int H = sp.H, W = sp.W, HW = H * W;
    const int tid = threadIdx.x;

    float il = 0.0f, op = 0.0f, cl = 0.0f;
    if (tid < NUM_A * T_TGT) {
        const int a = tid / T_TGT;
        const int t = tid % T_TGT;

        float x1 = targets[t * 5 + 0] / sp.red;
        float y1 = targets[t * 5 + 1] / sp.red;
        float x2 = targets[t * 5 + 2] / sp.red;
        float y2 = targets[t * 5 + 3] / sp.red;
        int   b  = (int)targets[t * 5 + 4];

        float tw = x2 - x1, th = y2 - y1, ta = tw * th;
        float iou3[3]; float best = -1.0f; int bi = 0;
        #pragma unroll
        for (int k = 0; k < 3; k++) {
            float inter = fminf(sp.aw[k], tw) * fminf(sp.ah[k], th);
            float iu = inter / (sp.aw[k] * sp.ah[k] + ta - inter);
            iou3[k] = iu;
            if (iu > best) { best = iu; bi = k; }
        }
        bool match = (iou3[a] > IOU_T) || (a == bi);

        if (match) {
            float cx = (x1 + x2) * 0.5f, cy = (y1 + y2) * 0.5f;
            float tx = cx - floorf(cx), ty = cy - floorf(cy);
            int ti = (int)floorf(cx); ti = min(max(ti, 0), W - 1);
            int tj = (int)floorf(cy); tj = min(max(tj, 0), H - 1);

            const float* p = pred + (b * 255 + a * 85) * HW + tj * W + ti;
            const float off = 0.5f * (sp.xys - 1.0f);
            float bx = sigm(p[0])      * sp.xys - off;   // no grid offset for CIoU
            float by = sigm(p[1 * HW]) * sp.xys - off;
            float bw = __expf(p[2 * HW]) * sp.aw[a];
            float bh = __expf(p[3 * HW]) * sp.ah[a];

            // CIoU(b1=pred, b2=target-box(tx,ty,tw,th))
            float b1x1 = bx - bw * 0.5f, b1x2 = bx + bw * 0.5f;
            float b1y1 = by - bh * 0.5f, b1y2 = by + bh * 0.5f;
            float b2x1 = tx - tw * 0.5f, b2x2 = tx + tw * 0.5f;
            float b2y1 = ty - th * 0.5f, b2y2 = ty + th * 0.5f;
            float ix = fmaxf(fminf(b1x2, b2x2) - fmaxf(b1x1, b2x1), 0.0f);
            float iy = fmaxf(fminf(b1y2, b2y2) - fmaxf(b1y1, b2y1), 0.0f);
            float inter = ix * iy;
            float uni = bw * bh + tw * th + EPSF - inter;
            float iou = inter / uni;
            float cw = fmaxf(b1x2, b2x2) - fminf(b1x1, b2x1);
            float chh = fmaxf(b1y2, b2y2) - fminf(b1y1, b2y1);
            float c2 = cw * cw + chh * chh + EPSF;
            float rho2 = (bx - tx) * (bx - tx) + (by - ty) * (by - ty);
            float dat = atanf(tw / th) - atanf(bw / bh);
            float v = (4.0f / (float)(M_PI * M_PI)) * dat * dat;
            float alpha = v / (1.0f - iou + v + EPSF);
            float ci = iou - (rho2 / c2 + v * alpha);
            il = 1.0f - ci;

            op = nslog(sigm(p[4 * HW]));

            int gtc = cats[t] - 1;
            float cls = 0.0f;
            for (int c = 0; c < NC; c++) {
                float cp = sigm(p[(5 + c) * HW]);
                cls += (c == gtc) ? nslog(cp) : nslog(1.0f - cp);
            }
            cl = cls;
        }
    }

    __shared__ float r0[256], r1[256], r2[256];
    r0[tid] = il; r1[tid] = op; r2[tid] = cl;
    __syncthreads();
    for (int s = 128; s > 0; s >>= 1) {
        if (tid < s) {
            r0[tid] += r0[tid + s];
            r1[tid] += r1[tid + s];
            r2[tid] += r2[tid + s];
        }
        __syncthreads();
    }
    if (tid == 0) {
        atomicAdd(&out[0], r0[0]);
        atomicAdd(&out[1], r1[0]);
        atomicAdd(&out[2], r2[0]);
    }
}

extern "C" void kernel_launch(void* const* d_in, const int* in_sizes, int n_in,
                              void* d_out, int out_size, void* d_ws, size_t ws_size,
                              hipStream_t stream) {
    const float* pred[3] = { (const float*)d_in[0],
                             (const float*)d_in[1],
                             (const float*)d_in[2] };
    const float* targets = (const float*)d_in[3];
    const int*   cats    = (const int*)d_in[4];
    float*       out     = (float*)d_out;

    hipMemsetAsync(out, 0, 3 * sizeof(float), stream);

    ScaleP sp[3];
    sp[0] = {76, 76,  8.0f, 1.20f,
             {12.f/8,  19.f/8,  40.f/8 }, {16.f/8,  36.f/8,  28.f/8 }};
    sp[1] = {38, 38, 16.0f, 1.10f,
             {36.f/16, 76.f/16, 72.f/16}, {75.f/16, 55.f/16, 146.f/16}};
    sp[2] = {19, 19, 32.0f, 1.05f,
             {142.f/32, 192.f/32, 459.f/32}, {110.f/32, 243.f/32, 401.f/32}};

    for (int i = 0; i < 3; i++) {
        const int N = NUM_B * NUM_A * sp[i].H * sp[i].W;
        const int grid = (N + 255) / 256;
        yolo_negobj_kernel<<<grid, 256, 0, stream>>>(pred[i], targets, out, sp[i]);
        yolo_pos_kernel<<<1, 256, 0, stream>>>(pred[i], targets, cats, out, sp[i]);
    }
}